// GNNEdgeHead_40338332844815
// MI455X (gfx1250) — compile-verified
//
#include <hip/hip_runtime.h>

typedef float v2f __attribute__((ext_vector_type(2)));
typedef float v8f __attribute__((ext_vector_type(8)));

// Problem constants (from the reference): D=64 features, K=8 negatives.
constexpr int D          = 64;
constexpr int K_NEG      = 8;
constexpr int SLOTS      = 9;     // 1 positive + 8 negatives
constexpr int EDGES_PER_WAVE  = 16;
constexpr int WAVES_PER_BLOCK = 8; // 256 threads = 8 wave32

// One wave32 handles 16 edges.
//   A (16x64 fp32)  = user rows, kept resident in 32 VGPRs (v2f a[16])
//   For each of 9 item slots: B (64x16 fp32) = item rows, 16-step chain of
//   V_WMMA_F32_16X16X4_F32 accumulating into an 8-VGPR f32 tile; keep diagonal.
__global__ __launch_bounds__(256)
void gnn_edge_head_wmma(const float* __restrict__ nf,    // [N, 64] fp32
                        const int*   __restrict__ eli,   // [2, E] (src row, dst row)
                        const int*   __restrict__ negi,  // [E, 8]
                        const int*   __restrict__ elab,  // [E]
                        float*       __restrict__ out,   // [E*9] preds ++ [E] labels
                        int E)
{
    const int lane = threadIdx.x & 31;
    const int wave = threadIdx.x >> 5;
    const int tile = blockIdx.x * WAVES_PER_BLOCK + wave;
    const int e0   = tile * EDGES_PER_WAVE;
    if (e0 >= E) return;

    const int m    = lane & 15;   // edge-in-tile this lane serves (M for A, N for B)
    const int half = lane >> 4;   // 0: K pair {0,1}; 1: K pair {2,3} within each step
    const int e    = e0 + m;

    // ---- Load A: user rows in WMMA 16x4 fp32 layout, full K=64 in registers ----
    const int    usrc = eli[e];                       // row 0 of edge_label_index
    const float* urow = nf + (size_t)usrc * D + 2 * half;
    v2f a[16];
    #pragma unroll
    for (int k = 0; k < 16; ++k)
        a[k] = *(const v2f*)(urow + 4 * k);

    // ---- Label passthrough (second tuple output), written as float ----
    if (half == 0)
        out[(size_t)E * SLOTS + e] = (float)elab[e];

    // ---- 9 item slots: slot 0 = positive item, slots 1..8 = negatives ----
    for (int s = 0; s < SLOTS; ++s) {
        int idx;
        if (s == 0) idx = eli[E + e];                          // dst (positive)
        else        idx = negi[(size_t)e * K_NEG + (s - 1)];   // negative s-1

        const float* brow = nf + (size_t)idx * D + 2 * half;
        v2f b[16];
        #pragma unroll
        for (int k = 0; k < 16; ++k)
            b[k] = *(const v2f*)(brow + 4 * k);

        // D = A x B + C over K=64 via 16 chained 16x16x4 fp32 WMMAs
        v8f c = {};
        #pragma unroll
        for (int k = 0; k < 16; ++k)
            c = __builtin_amdgcn_wmma_f32_16x16x4_f32(
                    /*neg_a=*/false, a[k],
                    /*neg_b=*/false, b[k],
                    /*c_mod=*/(short)0, c,
                    /*reuse_a=*/false, /*reuse_b=*/false);

        // Diagonal extraction per D layout:
        //   edge r (r<8)  -> VGPR r,   lane r
        //   edge r+8      -> VGPR r,   lane r+24
        float dval = c[0];
        #pragma unroll
        for (int r = 1; r < 8; ++r)
            if ((lane & 7) == r) dval = c[r];

        const int  edge_in_tile = (lane < 8) ? lane : (lane - 16);
        const bool writer       = (lane < 8) || (lane >= 24);
        if (writer)
            out[(size_t)(e0 + edge_in_tile) * SLOTS + s] = dval;
    }
}

extern "C" void kernel_launch(void* const* d_in, const int* in_sizes, int n_in,
                              void* d_out, int out_size, void* d_ws, size_t ws_size,
                              hipStream_t stream) {
    const float* nf   = (const float*)d_in[0];   // node_feature  [N,64] fp32
    const int*   eli  = (const int*)  d_in[1];   // edge_label_index [2,E]
    const int*   negi = (const int*)  d_in[2];   // neg_index [E,8]
    const int*   elab = (const int*)  d_in[3];   // edge_label [E]
    float*       out  = (float*)d_out;

    const int E      = in_sizes[3];                       // 262144
    const int tiles  = (E + EDGES_PER_WAVE - 1) / EDGES_PER_WAVE;
    const int blocks = (tiles + WAVES_PER_BLOCK - 1) / WAVES_PER_BLOCK;

    gnn_edge_head_wmma<<<dim3(blocks), dim3(256), 0, stream>>>(
        nf, eli, negi, elab, out, E);
}